// YOLODetector_36507222016094
// MI455X (gfx1250) — compile-verified
//
#include <hip/hip_runtime.h>
#include <stdint.h>

#define BATCH   16
#define NANCH   25200
#define NATTR   85
#define NCLS    80
#define KCAND   1024
#define MAXDET  300
#define NCHUNK  32
#define NPAD    (NCHUNK * KCAND)   /* 32768 padded slots per batch */
#define TROWS   128                /* rows per score tile (multiple of 4) */
#define NTILE   (NPAD / TROWS)     /* 256 tiles per batch */
#define CONF_T  0.4f
#define IOU_T   0.25f

#define VALID_OFF (BATCH * MAXDET * 6)            /* 28800 */
#define FEATS_OFF (VALID_OFF + BATCH * MAXDET)    /* 33600 */

typedef unsigned long long u64;
typedef unsigned int       u32;
typedef int v4i __attribute__((vector_size(16)));

/* ---- gfx1250 async global->LDS path (guarded; falls back to plain LDS) ---- */
#if defined(__has_builtin)
#  if __has_builtin(__builtin_amdgcn_global_load_async_to_lds_b128) && \
      __has_builtin(__builtin_amdgcn_s_wait_asynccnt)
#    define USE_ASYNC_LDS 1
#  endif
#endif
#ifndef USE_ASYNC_LDS
#  define USE_ASYNC_LDS 0
#endif

__device__ __forceinline__ void async_copy16(const void* g, void* l) {
#if USE_ASYNC_LDS
  __builtin_amdgcn_global_load_async_to_lds_b128(
      (__attribute__((address_space(1))) v4i*)(void*)g,
      (__attribute__((address_space(3))) v4i*)l, 0, 0);
#else
  *(v4i*)l = *(const v4i*)g;
#endif
}
__device__ __forceinline__ void async_wait0() {
#if USE_ASYNC_LDS
  __builtin_amdgcn_s_wait_asynccnt(0);
#endif
}

__device__ __forceinline__ u32 f2ord(float f) {
  u32 u = __float_as_uint(f);
  return (u & 0x80000000u) ? ~u : (u | 0x80000000u);
}
__device__ __forceinline__ float ord2f(u32 u) {
  return (u & 0x80000000u) ? __uint_as_float(u ^ 0x80000000u)
                           : __uint_as_float(~u);
}

/* Pass 1: stream pred once (the only roofline-relevant traffic: ~137MB).
   Stage 128-row x 340B tiles into LDS with async b128 copies (coalesced,
   ASYNCcnt-tracked), then per-thread 80-class max out of LDS (stride 85
   words -> conflict-free across 64 banks).
   key = ordered(masked_conf)<<32 | (0xFFFFFFFF - n): desc score, asc index
   (exact lax.top_k tie-break). Padded slots (n >= NANCH) get key 0. */
__global__ void __launch_bounds__(TROWS) k_score(const float* __restrict__ pred,
                                                 u64* __restrict__ keys) {
  __shared__ __align__(16) float tile[TROWS * NATTR];   /* 43520 B */
  int b = blockIdx.x >> 8;              /* NTILE == 256 tiles per batch */
  int tileIdx = blockIdx.x & (NTILE - 1);
  int t = threadIdx.x;
  int rowBase = tileIdx * TROWS;
  int nrows = NANCH - rowBase;
  if (nrows > TROWS) nrows = TROWS;
  if (nrows > 0) {
    /* nrows is 128 or 112 -> nrows*NATTR is a multiple of 4 -> whole tile
       moves as 16B packets; global/LDS bases are 16B aligned. */
    int nvec = (nrows * NATTR) >> 2;
    const char* gbase =
        (const char*)(pred + ((size_t)b * NANCH + rowBase) * NATTR);
    for (int idx = t; idx < nvec; idx += TROWS)
      async_copy16(gbase + (size_t)idx * 16, (char*)tile + (size_t)idx * 16);
    async_wait0();
  }
  __syncthreads();
  int n = rowBase + t;
  u64 key = 0ull;
  if (n < NANCH) {
    const float* row = tile + t * NATTR;
    float obj  = row[4];
    float best = -1e30f;
#pragma unroll 10
    for (int c = 0; c < NCLS; ++c) {
      float v = row[5 + c] * obj;
      if (v > best) best = v;
    }
    float masked = (obj > CONF_T && best > CONF_T) ? best : -1.0f;
    key = ((u64)f2ord(masked) << 32) | (u64)(0xFFFFFFFFu - (u32)n);
  }
  keys[(size_t)b * NPAD + n] = key;
}

/* Pass 2: bitonic sort each 1024-key chunk descending (512 thr, LDS 8KB). */
__global__ void __launch_bounds__(512) k_chunk_sort(u64* __restrict__ keys) {
  __shared__ u64 s[KCAND];
  int b = blockIdx.x / NCHUNK, c = blockIdx.x - b * NCHUNK, t = threadIdx.x;
  u64* base = keys + (size_t)b * NPAD + (size_t)c * KCAND;
  s[t] = base[t];
  s[t + 512] = base[t + 512];
  for (int k2 = 2; k2 <= KCAND; k2 <<= 1)
    for (int j = k2 >> 1; j > 0; j >>= 1) {
      __syncthreads();
      int i = ((t & ~(j - 1)) << 1) | (t & (j - 1));
      int p = i | j;
      u64 a = s[i], bb = s[p];
      bool up = ((i & k2) == 0);                /* descending subsequence */
      if (up ? (a < bb) : (a > bb)) { s[i] = bb; s[p] = a; }
    }
  __syncthreads();
  base[t] = s[t];
  base[t + 512] = s[t + 512];
}

/* Pass 3 (x5 rounds): tournament merge. Load A desc + B reversed (asc) ->
   bitonic 2048; merge keeping max-first; write back top 1024 to slot A. */
__global__ void __launch_bounds__(KCAND) k_merge(u64* __restrict__ keys,
                                                 int r, int pairs) {
  __shared__ u64 m[2 * KCAND];
  int b = blockIdx.x / pairs, p = blockIdx.x - b * pairs, t = threadIdx.x;
  size_t baseA = (size_t)b * NPAD + ((size_t)(p << (r + 1))) * KCAND;
  size_t baseB = baseA + ((size_t)(1 << r)) * KCAND;
  m[t]         = keys[baseA + t];
  m[KCAND + t] = keys[baseB + (KCAND - 1 - t)];
  for (int j = KCAND; j > 0; j >>= 1) {
    __syncthreads();
    int i = ((t & ~(j - 1)) << 1) | (t & (j - 1));
    int q = i | j;
    u64 a = m[i], bb = m[q];
    if (a < bb) { m[i] = bb; m[q] = a; }
  }
  __syncthreads();
  keys[baseA + t] = m[t];
}

/* Pass 4: decode top-1024, gather boxes (xywh->xyxy), argmax class, valid0. */
__global__ void k_gather(const float* __restrict__ pred,
                         const u64* __restrict__ keys,
                         float* __restrict__ boxes, float* __restrict__ scoreK,
                         int* __restrict__ clsK, u32* __restrict__ keep) {
  int gid = blockIdx.x * blockDim.x + threadIdx.x;   /* BATCH*KCAND */
  int b = gid / KCAND;
  u64 key = keys[(size_t)b * NPAD + (gid - b * KCAND)];
  float sc = ord2f((u32)(key >> 32));
  float x1 = 0.f, y1 = 0.f, x2 = 0.f, y2 = 0.f;
  int cls = 0;
  u32 kp = 0;
  if (sc > CONF_T) {                       /* only these rows ever matter */
    u32 n = 0xFFFFFFFFu - (u32)(key & 0xFFFFFFFFull);
    const float* row = pred + ((size_t)b * NANCH + n) * NATTR;
    float cx = row[0], cy = row[1], w = row[2], h = row[3];
    x1 = cx - w * 0.5f; y1 = cy - h * 0.5f;
    x2 = cx + w * 0.5f; y2 = cy + h * 0.5f;
    float obj = row[4], best = -1e30f;
    for (int c = 0; c < NCLS; ++c) {
      float v = row[5 + c] * obj;
      if (v > best) { best = v; cls = c; }   /* first-occurrence argmax */
    }
    kp = 1;
  }
  ((float4*)boxes)[gid] = make_float4(x1, y1, x2, y2);
  scoreK[gid] = sc;
  clsK[gid]   = cls;
  keep[gid]   = kp;
}

/* Pass 5: greedy NMS, one block per batch, everything in LDS (24KB).
   Boxes staged via async global->LDS. Exact reference semantics:
   sequential over i, suppress j>i with IoU>0.25 (strict), clip(wh,0),
   denom + 1e-9. */
__global__ void __launch_bounds__(KCAND) k_nms(const float* __restrict__ boxes,
                                               u32* __restrict__ keep) {
  __shared__ __align__(16) float4 sb[KCAND];
  __shared__ float sa[KCAND];
  __shared__ u32   sk[KCAND];
  int b = blockIdx.x, t = threadIdx.x;
  const float4* src = ((const float4*)boxes) + (size_t)b * KCAND + t;
  async_copy16(src, &sb[t]);
  async_wait0();
  float4 v = sb[t];
  sa[t] = (v.z - v.x) * (v.w - v.y);
  sk[t] = keep[(size_t)b * KCAND + t];
  __syncthreads();
  for (int i = 0; i < KCAND; ++i) {
    if (sk[i] && t > i && sk[t]) {
      float4 bi = sb[i];
      float xx1 = fmaxf(bi.x, v.x), yy1 = fmaxf(bi.y, v.y);
      float xx2 = fminf(bi.z, v.z), yy2 = fminf(bi.w, v.w);
      float w = fmaxf(xx2 - xx1, 0.f), h = fmaxf(yy2 - yy1, 0.f);
      float inter = w * h;
      float iou = inter / (sa[i] + sa[t] - inter + 1e-9f);
      if (iou > IOU_T) sk[t] = 0;
    }
    __syncthreads();
  }
  keep[(size_t)b * KCAND + t] = sk[t];
}

/* Pass 6: stable partition via block scan, emit dets/valid/feats. */
__global__ void __launch_bounds__(KCAND) k_final(
    const float* __restrict__ boxes, const float* __restrict__ scoreK,
    const int* __restrict__ clsK, const u32* __restrict__ keep,
    const int* __restrict__ img, float* __restrict__ out) {
  __shared__ u32 kp[KCAND];
  __shared__ int scn[KCAND];
  __shared__ int order[KCAND];
  int b = blockIdx.x, t = threadIdx.x;
  u32 k = keep[(size_t)b * KCAND + t];
  kp[t] = k;
  scn[t] = (int)k;
  __syncthreads();
  for (int off = 1; off < KCAND; off <<= 1) {      /* inclusive scan */
    int add = (t >= off) ? scn[t - off] : 0;
    __syncthreads();
    scn[t] += add;
    __syncthreads();
  }
  int total = scn[KCAND - 1];
  int pos = k ? (scn[t] - 1) : (total + (t - scn[t]));
  order[pos] = t;
  __syncthreads();
  if (t < MAXDET) {
    int s = order[t];
    u32 kv = kp[s];
    size_t g = (size_t)b * KCAND + s;
    float x1 = 0.f, y1 = 0.f, x2 = 0.f, y2 = 0.f, sv = 0.f, cv = 0.f;
    int ci = 0;
    if (kv) {
      float4 bb = ((const float4*)boxes)[g];
      x1 = bb.x; y1 = bb.y; x2 = bb.z; y2 = bb.w;
      sv = scoreK[g];
      ci = clsK[g];
      cv = (float)ci;
    }
    float* d = out + ((size_t)b * MAXDET + t) * 6;
    d[0] = x1; d[1] = y1; d[2] = x2; d[3] = y2; d[4] = sv; d[5] = cv;
    out[VALID_OFF + (size_t)b * MAXDET + t] = kv ? 1.0f : 0.0f;
    if (b == 0) {
      float* f = out + FEATS_OFF + (size_t)t * (2 + NCLS);
      if (kv) {
        float scale = 1.0f / (float)img[0];
        f[0] = (x1 + x2) * 0.5f * scale;
        f[1] = (y1 + y2) * 0.5f * scale;
        for (int c = 0; c < NCLS; ++c) f[2 + c] = 0.0f;
        f[2 + ci] = 1.0f;
      } else {
        for (int c = 0; c < 2 + NCLS; ++c) f[c] = 0.0f;
      }
    }
  }
}

extern "C" void kernel_launch(void* const* d_in, const int* in_sizes, int n_in,
                              void* d_out, int out_size, void* d_ws,
                              size_t ws_size, hipStream_t stream) {
  (void)in_sizes; (void)n_in; (void)out_size; (void)ws_size;
  const float* pred = (const float*)d_in[0];
  const int*   img  = (const int*)d_in[1];
  float*       out  = (float*)d_out;

  char* ws = (char*)d_ws;
  size_t off = 0;
  auto alloc = [&](size_t bytes) -> void* {
    void* p = ws + off;
    off += (bytes + 255) & ~(size_t)255;
    return p;
  };
  u64*   keys   = (u64*)alloc((size_t)BATCH * NPAD * sizeof(u64));     /* 4 MB */
  float* boxes  = (float*)alloc((size_t)BATCH * KCAND * 4 * 4);        /* 256K */
  float* scoreK = (float*)alloc((size_t)BATCH * KCAND * 4);
  int*   clsK   = (int*)alloc((size_t)BATCH * KCAND * 4);
  u32*   keep   = (u32*)alloc((size_t)BATCH * KCAND * 4);

  k_score<<<BATCH * NTILE, TROWS, 0, stream>>>(pred, keys);
  k_chunk_sort<<<BATCH * NCHUNK, 512, 0, stream>>>(keys);
  for (int r = 0; r < 5; ++r) {
    int pairs = NCHUNK >> (r + 1);
    k_merge<<<BATCH * pairs, KCAND, 0, stream>>>(keys, r, pairs);
  }
  k_gather<<<(BATCH * KCAND) / 256, 256, 0, stream>>>(pred, keys, boxes,
                                                      scoreK, clsK, keep);
  k_nms<<<BATCH, KCAND, 0, stream>>>(boxes, keep);
  k_final<<<BATCH, KCAND, 0, stream>>>(boxes, scoreK, clsK, keep, img, out);
}